// QuantumPolicy_41807211659318
// MI455X (gfx1250) — compile-verified
//
#include <hip/hip_runtime.h>
#include <hip/hip_bf16.h>
#include <math.h>

typedef __attribute__((ext_vector_type(2))) float v2f;
typedef __attribute__((ext_vector_type(8))) float v8f;

__device__ __forceinline__ v8f wmma_f32(v2f a, v2f b, v8f c) {
  // D = A(16x4) * B(4x16) + C, full fp32
  return __builtin_amdgcn_wmma_f32_16x16x4_f32(false, a, false, b, (short)0, c,
                                               false, false);
}

// ---------------------------------------------------------------------------
// Kernel A: build the fixed 64x64 circuit matrix.
// Thread t simulates circuit(e_t): result c[n] = U[n][t].
// Stored as Bmat[n*64 + t]  (i.e. Bmat[n][k] = B[k][n] with B[k][n]=U[n][k]),
// so GEMM B-fragments read CONTIGUOUS k,k+1 pairs.
// ---------------------------------------------------------------------------
__global__ __launch_bounds__(64) void build_unitary(
    const float* __restrict__ qw, float* __restrict__ Bmat) {
  const int t = threadIdx.x;  // 0..63 basis column
  float st[64];
#pragma unroll
  for (int k = 0; k < 64; ++k) st[k] = (k == t) ? 1.0f : 0.0f;

  for (int l = 0; l < 4; ++l) {
    // ring of CNOTs: ctrl=i, tgt=(i+1)%6 ; qubit q <-> bit (5-q)
#pragma unroll
    for (int i = 0; i < 6; ++i) {
      const int cm = 1 << (5 - i);
      const int tm = 1 << (5 - ((i + 1) % 6));
#pragma unroll
      for (int k = 0; k < 64; ++k) {
        if ((k & cm) && !(k & tm)) {
          float a = st[k];
          st[k] = st[k | tm];
          st[k | tm] = a;
        }
      }
    }
    // scalar RY on each qubit
#pragma unroll
    for (int i = 0; i < 6; ++i) {
      const float th = qw[l * 6 + i] * 0.5f;
      const float c = cosf(th), s = sinf(th);
      const int qm = 1 << (5 - i);
#pragma unroll
      for (int k = 0; k < 64; ++k) {
        if (!(k & qm)) {
          float a0 = st[k], a1 = st[k | qm];
          st[k] = c * a0 - s * a1;      // new0 =  c*a0 - s*a1
          st[k | qm] = s * a0 + c * a1; // new1 =  s*a0 + c*a1
        }
      }
    }
  }
#pragma unroll
  for (int n = 0; n < 64; ++n) Bmat[n * 64 + t] = st[n];
}

// ---------------------------------------------------------------------------
// Kernel B: full pipeline. 8 waves/block, 16 samples per wave.
// ---------------------------------------------------------------------------
__global__ __launch_bounds__(256) void qpolicy_main(
    const float* __restrict__ x, const float* __restrict__ W1,
    const float* __restrict__ b1, const float* __restrict__ W2,
    const float* __restrict__ b2, const float* __restrict__ Wp,
    const float* __restrict__ bp, const float* __restrict__ Wv,
    const float* __restrict__ bv, const float* __restrict__ Umat,
    float* __restrict__ out, int nB) {
  __shared__ float Ulds[64 * 66];       // [n][k], padded stride 66
  __shared__ float Htile[8][16 * 66];   // per-wave 16 x 66 staging

  // cooperative stage of U into LDS
  for (int idx = threadIdx.x; idx < 4096; idx += 256) {
    int n = idx >> 6, k = idx & 63;
    Ulds[n * 66 + k] = Umat[idx];
  }
  __syncthreads();

  const int lane = threadIdx.x & 31;
  const int wv = threadIdx.x >> 5;
  const int half = lane >> 4;  // 0/1 : selects K/M sub-block per ISA layout
  const int lm = lane & 15;
  const long row0 = ((long)blockIdx.x * 8 + wv) * 16;
  if (row0 >= nB) return;
  float* H = &Htile[wv][0];

  // ---- loop-invariant B fragments ------------------------------------
  // GEMM1: B[k][n] = W1[n][k] (W1 is (64,6) row-major), K padded 6->8
  v2f w1f[2][4];
#pragma unroll
  for (int kb = 0; kb < 2; ++kb)
#pragma unroll
    for (int nt = 0; nt < 4; ++nt) {
      int n = nt * 16 + lm;
      int k = kb * 4 + half * 2;
      w1f[kb][nt].x = (k < 6) ? W1[n * 6 + k] : 0.0f;
      w1f[kb][nt].y = (k + 1 < 6) ? W1[n * 6 + k + 1] : 0.0f;
    }
  // GEMM2: B[k][n] = W2[n][k] (W2 is (6,64) row-major), N padded 6->16
  v2f w2f[16];
  {
    const bool nv = (lm < 6);
#pragma unroll
    for (int kb = 0; kb < 16; ++kb) {
      int k = kb * 4 + half * 2;
      w2f[kb].x = nv ? W2[lm * 64 + k] : 0.0f;
      w2f[kb].y = nv ? W2[lm * 64 + k + 1] : 0.0f;
    }
  }
  float b1v[4];
#pragma unroll
  for (int nt = 0; nt < 4; ++nt) b1v[nt] = b1[nt * 16 + lm];
  const float b2v = (lm < 6) ? b2[lm] : 0.0f;

  // ---- GEMM1: h = relu(x @ W1^T + b1) --------------------------------
  v2f xf[2];
  {
    const float* xr = x + (row0 + lm) * 6;  // A: M=lm, K=2*half+v
    int k0 = half * 2;
    xf[0].x = xr[k0];
    xf[0].y = xr[k0 + 1];
    int k1 = 4 + half * 2;  // 4,5 valid; 6,7 -> zero pad
    xf[1].x = (k1 < 6) ? xr[k1] : 0.0f;
    xf[1].y = (k1 + 1 < 6) ? xr[k1 + 1] : 0.0f;
  }
#pragma unroll
  for (int nt = 0; nt < 4; ++nt) {
    v8f acc = {0.f, 0.f, 0.f, 0.f, 0.f, 0.f, 0.f, 0.f};
    acc = wmma_f32(xf[0], w1f[0][nt], acc);
    acc = wmma_f32(xf[1], w1f[1][nt], acc);
#pragma unroll
    for (int v = 0; v < 8; ++v) {  // D: M=v+8*half, N=nt*16+lm
      float hv = fmaxf(acc[v] + b1v[nt], 0.0f);
      H[(v + half * 8) * 66 + nt * 16 + lm] = hv;
    }
  }

  // ---- GEMM2: xp = relu(h @ W2^T + b2) -------------------------------
  v8f acc2 = {0.f, 0.f, 0.f, 0.f, 0.f, 0.f, 0.f, 0.f};
#pragma unroll
  for (int kb = 0; kb < 16; ++kb) {
    v2f a = *(const v2f*)(H + lm * 66 + kb * 4 + half * 2);
    acc2 = wmma_f32(a, w2f[kb], acc2);
  }
  if (lm < 6) {
#pragma unroll
    for (int v = 0; v < 8; ++v)
      H[(v + half * 8) * 66 + lm] = fmaxf(acc2[v] + b2v, 0.0f);
  }

  // ---- build product state S0 for sample m = lm ----------------------
  float cq[6], sq[6];
#pragma unroll
  for (int q = 0; q < 6; ++q) {
    float hh = 0.5f * H[lm * 66 + q];
    cq[q] = cosf(hh);
    sq[q] = sinf(hh);
  }
  float A01[4], B23[4], C45[4];
#pragma unroll
  for (int b = 0; b < 4; ++b) {
    A01[b] = ((b & 2) ? sq[0] : cq[0]) * ((b & 1) ? sq[1] : cq[1]);
    B23[b] = ((b & 2) ? sq[2] : cq[2]) * ((b & 1) ? sq[3] : cq[3]);
    C45[b] = ((b & 2) ? sq[4] : cq[4]) * ((b & 1) ? sq[5] : cq[5]);
  }
  v2f s0f[16];  // A frags: lane holds amp(k), amp(k+1), k = kb*4 + 2*half
#pragma unroll
  for (int kb = 0; kb < 16; ++kb) {
    int k = kb * 4 + half * 2;
    s0f[kb].x = A01[(k >> 4) & 3] * B23[(k >> 2) & 3] * C45[k & 3];
    int k1 = k + 1;
    s0f[kb].y = A01[(k1 >> 4) & 3] * B23[(k1 >> 2) & 3] * C45[k1 & 3];
  }

  // ---- GEMM3: S1 = S0 @ U^T  (the heavy part: 64 WMMAs) --------------
  v8f acc3[4];
#pragma unroll
  for (int nt = 0; nt < 4; ++nt) {
    v8f acc = {0.f, 0.f, 0.f, 0.f, 0.f, 0.f, 0.f, 0.f};
#pragma unroll
    for (int kb = 0; kb < 16; ++kb) {
      v2f b = *(const v2f*)(Ulds + (nt * 16 + lm) * 66 + kb * 4 + half * 2);
      acc = wmma_f32(s0f[kb], b, acc);
    }
    acc3[nt] = acc;
  }
#pragma unroll
  for (int nt = 0; nt < 4; ++nt)
#pragma unroll
    for (int v = 0; v < 8; ++v)
      H[(v + half * 8) * 66 + nt * 16 + lm] = acc3[nt][v];

  // ---- Z expectations: qout[q] = sum_n sgn(q,n) * S1[n]^2 ------------
  float qacc[6] = {0.f, 0.f, 0.f, 0.f, 0.f, 0.f};
#pragma unroll
  for (int j = 0; j < 32; ++j) {
    int n = half * 32 + j;
    float sv = H[lm * 66 + n];
    float p = sv * sv;
#pragma unroll
    for (int q = 0; q < 6; ++q)
      qacc[q] += ((n >> (5 - q)) & 1) ? -p : p;
  }
#pragma unroll
  for (int q = 0; q < 6; ++q) qacc[q] += __shfl_xor(qacc[q], 16, 32);

  // ---- tail: softmax policy + value (lanes 0..15, one sample each) ---
  if (half == 0) {
    const long srow = row0 + lm;
    float lg[4];
#pragma unroll
    for (int a = 0; a < 4; ++a) {
      float s = bp[a];
#pragma unroll
      for (int q = 0; q < 6; ++q) s += Wp[a * 6 + q] * qacc[q];
      lg[a] = s;
    }
    float mx = fmaxf(fmaxf(lg[0], lg[1]), fmaxf(lg[2], lg[3]));
    float e[4], sum = 0.0f;
#pragma unroll
    for (int a = 0; a < 4; ++a) {
      e[a] = expf(lg[a] - mx);
      sum += e[a];
    }
    float inv = 1.0f / sum;
#pragma unroll
    for (int a = 0; a < 4; ++a) out[srow * 4 + a] = e[a] * inv;
    float val = bv[0];
#pragma unroll
    for (int q = 0; q < 6; ++q) val += Wv[q] * qacc[q];
    out[(long)nB * 4 + srow] = val;
  }
}

extern "C" void kernel_launch(void* const* d_in, const int* in_sizes, int n_in,
                              void* d_out, int out_size, void* d_ws,
                              size_t ws_size, hipStream_t stream) {
  const float* x  = (const float*)d_in[0];
  const float* W1 = (const float*)d_in[1];
  const float* b1 = (const float*)d_in[2];
  const float* W2 = (const float*)d_in[3];
  const float* b2 = (const float*)d_in[4];
  const float* qw = (const float*)d_in[5];
  const float* Wp = (const float*)d_in[6];
  const float* bp = (const float*)d_in[7];
  const float* Wv = (const float*)d_in[8];
  const float* bv = (const float*)d_in[9];
  float* out = (float*)d_out;
  float* Bmat = (float*)d_ws;  // 64*64 floats = 16 KB

  const int nB = in_sizes[0] / 6;

  build_unitary<<<1, 64, 0, stream>>>(qw, Bmat);

  const int samplesPerBlock = 128;  // 8 waves * 16 samples
  const int grid = (nB + samplesPerBlock - 1) / samplesPerBlock;
  qpolicy_main<<<grid, 256, 0, stream>>>(x, W1, b1, W2, b2, Wp, bp, Wv, bv,
                                         Bmat, out, nB);
}